// GNNStack_13615046328861
// MI455X (gfx1250) — compile-verified
//
#include <hip/hip_runtime.h>
#include <hip/hip_bf16.h>
#include <math.h>

typedef __attribute__((ext_vector_type(2))) float v2f;
typedef __attribute__((ext_vector_type(4))) float v4f;
typedef __attribute__((ext_vector_type(8))) float v8f;
typedef __attribute__((ext_vector_type(4))) unsigned v4u;

// ---- order-preserving float<->uint encoding for atomicMax on floats ----
__device__ __forceinline__ unsigned fenc(float f) {
  unsigned u = __float_as_uint(f);
  return (u & 0x80000000u) ? ~u : (u | 0x80000000u);
}
__device__ __forceinline__ float fdec(unsigned u) {
  unsigned v = (u & 0x80000000u) ? (u & 0x7fffffffu) : ~u;
  return __uint_as_float(v);
}
__device__ __forceinline__ float lrelu(float x) { return x > 0.f ? x : 0.2f * x; }

// =====================================================================
// WMMA GEMM:  Y[M x OUT] = X[M x K] @ W[OUT x K]^T + bias   (K <= 128)
// One block per 16-row tile; wave w owns output-column tile w.
// X tile staged once in LDS (padded stride 132 -> no bank conflicts on
// the A-fragment reads). fp32 WMMA 16x16x4 accumulation.
// A frag: lane l supplies row (l&15), K slice 2*(l>>4)+{0,1}
// B frag: lane l supplies col (l&15) (a row of W), same K slice
// D:      lane l, vgpr r -> M = r + 8*(l>>4), N = l&15
// =====================================================================
#define XS_STRIDE 132
__global__ void wmma_gemm_lds(const float* __restrict__ X,
                              const float* __restrict__ W,
                              const float* __restrict__ bias,
                              float* __restrict__ Y,
                              int M, int K, int OUT) {
  __shared__ float xs[16 * XS_STRIDE];
  const int wave = threadIdx.x >> 5;
  const int lane = threadIdx.x & 31;
  const int mtile = blockIdx.x;

  // cooperative, coalesced stage of the 16 x K X-tile into LDS
  for (int i = threadIdx.x; i < 16 * K; i += blockDim.x) {
    int rr = i / K;
    int kk = i - rr * K;
    int row = mtile * 16 + rr;
    row = row < M ? row : (M - 1);
    xs[rr * XS_STRIDE + kk] = X[(size_t)row * K + kk];
  }
  __syncthreads();

  const int r = lane & 15;
  const int half = lane >> 4;
  const int bcol = wave * 16 + r;
  const bool colok = bcol < OUT;
  const float* wp = colok ? (W + (size_t)bcol * K + 2 * half) : (W + 2 * half);
  const float* xp = xs + r * XS_STRIDE + 2 * half;

  v8f acc = {0.f, 0.f, 0.f, 0.f, 0.f, 0.f, 0.f, 0.f};
  for (int k = 0; k < K; k += 4) {
    v2f a;
    a[0] = xp[k];
    a[1] = xp[k + 1];
    v2f b;
    b[0] = wp[k];
    b[1] = wp[k + 1];
    acc = __builtin_amdgcn_wmma_f32_16x16x4_f32(false, a, false, b,
                                                (short)0, acc, false, false);
  }
  if (!colok) return;
  float bv = bias ? bias[bcol] : 0.f;
#pragma unroll
  for (int vr = 0; vr < 8; ++vr) {
    int rowo = mtile * 16 + vr + 8 * half;
    if (rowo < M) Y[(size_t)rowo * OUT + bcol] = acc[vr] + bv;
  }
}

// ---------------------------------------------------------------------
// Per (node, head): attention scalars a_i (dst half), a_j (src half),
// and init segment-max with the self-loop logit.
// ---------------------------------------------------------------------
__global__ void attn_scalars(const float* __restrict__ Hb,
                             const float* __restrict__ att,
                             float* __restrict__ ai, float* __restrict__ aj,
                             unsigned* __restrict__ menc,
                             int Nn, int H, int C) {
  int idx = blockIdx.x * blockDim.x + threadIdx.x;
  if (idx >= Nn * H) return;
  int n = idx / H, h = idx - n * H;
  const float* hp = Hb + (size_t)n * H * C + (size_t)h * C;
  const float* ar = att + (size_t)h * 2 * C;
  float si = 0.f, sj = 0.f;
  for (int c = 0; c < C; ++c) {
    float v = hp[c];
    si += v * ar[c];
    sj += v * ar[C + c];
  }
  ai[idx] = si;
  aj[idx] = sj;
  menc[idx] = fenc(lrelu(si + sj));  // self-loop logit seeds the segment max
}

// ---------------------------------------------------------------------
// Layer-1 (H=8) edge pass 1: vectorized 8-head gather + atomicMax
// ---------------------------------------------------------------------
__global__ void edge_max8(const int* __restrict__ src, const int* __restrict__ dst,
                          const float* __restrict__ ai, const float* __restrict__ aj,
                          unsigned* __restrict__ menc, int E) {
  int e = blockIdx.x * blockDim.x + threadIdx.x;
  if (e >= E) return;
  int s = src[e], d = dst[e];
  v4f i0 = *(const v4f*)(ai + (size_t)d * 8);
  v4f i1 = *(const v4f*)(ai + (size_t)d * 8 + 4);
  v4f j0 = *(const v4f*)(aj + (size_t)s * 8);
  v4f j1 = *(const v4f*)(aj + (size_t)s * 8 + 4);
  unsigned* mp = menc + (size_t)d * 8;
#pragma unroll
  for (int h = 0; h < 4; ++h) atomicMax(&mp[h], fenc(lrelu(i0[h] + j0[h])));
#pragma unroll
  for (int h = 0; h < 4; ++h) atomicMax(&mp[4 + h], fenc(lrelu(i1[h] + j1[h])));
}

// ---------------------------------------------------------------------
// Generic edge pass 1 (used for layer 2, H=1)
// ---------------------------------------------------------------------
__global__ void edge_max(const int* __restrict__ src, const int* __restrict__ dst,
                         const float* __restrict__ ai, const float* __restrict__ aj,
                         unsigned* __restrict__ menc, int E, int H) {
  int e = blockIdx.x * blockDim.x + threadIdx.x;
  if (e >= E) return;
  int s = src[e], d = dst[e];
  for (int h = 0; h < H; ++h) {
    float ev = lrelu(ai[d * H + h] + aj[s * H + h]);
    atomicMax(&menc[d * H + h], fenc(ev));
  }
}

// ---------------------------------------------------------------------
// Node pass: denom initialized with the self-loop exp term
// ---------------------------------------------------------------------
__global__ void denom_init(const float* __restrict__ ai, const float* __restrict__ aj,
                           const unsigned* __restrict__ menc,
                           float* __restrict__ denom, int NH) {
  int idx = blockIdx.x * blockDim.x + threadIdx.x;
  if (idx >= NH) return;
  float es = lrelu(ai[idx] + aj[idx]);
  denom[idx] = __expf(es - fdec(menc[idx]));
}

// ---------------------------------------------------------------------
// Layer-1 (H=8) edge pass 2: vectorized gathers + atomicAdd of exps
// ---------------------------------------------------------------------
__global__ void edge_denom8(const int* __restrict__ src, const int* __restrict__ dst,
                            const float* __restrict__ ai, const float* __restrict__ aj,
                            const unsigned* __restrict__ menc,
                            float* __restrict__ denom, int E) {
  int e = blockIdx.x * blockDim.x + threadIdx.x;
  if (e >= E) return;
  int s = src[e], d = dst[e];
  v4f i0 = *(const v4f*)(ai + (size_t)d * 8);
  v4f i1 = *(const v4f*)(ai + (size_t)d * 8 + 4);
  v4f j0 = *(const v4f*)(aj + (size_t)s * 8);
  v4f j1 = *(const v4f*)(aj + (size_t)s * 8 + 4);
  v4u m0 = *(const v4u*)(menc + (size_t)d * 8);
  v4u m1 = *(const v4u*)(menc + (size_t)d * 8 + 4);
  float* dp = denom + (size_t)d * 8;
#pragma unroll
  for (int h = 0; h < 4; ++h)
    atomicAdd(&dp[h], __expf(lrelu(i0[h] + j0[h]) - fdec(m0[h])));
#pragma unroll
  for (int h = 0; h < 4; ++h)
    atomicAdd(&dp[4 + h], __expf(lrelu(i1[h] + j1[h]) - fdec(m1[h])));
}

// ---------------------------------------------------------------------
// Generic edge pass 2 (layer 2, H=1)
// ---------------------------------------------------------------------
__global__ void edge_denom(const int* __restrict__ src, const int* __restrict__ dst,
                           const float* __restrict__ ai, const float* __restrict__ aj,
                           const unsigned* __restrict__ menc,
                           float* __restrict__ denom, int E, int H) {
  int e = blockIdx.x * blockDim.x + threadIdx.x;
  if (e >= E) return;
  int s = src[e], d = dst[e];
  for (int h = 0; h < H; ++h) {
    float ev = lrelu(ai[d * H + h] + aj[s * H + h]);
    atomicAdd(&denom[d * H + h], __expf(ev - fdec(menc[d * H + h])));
  }
}

// ---------------------------------------------------------------------
// Node pass: out = h * alpha_self  (fully initializes accumulator)
// ---------------------------------------------------------------------
__global__ void out_init(const float* __restrict__ Hb,
                         const float* __restrict__ ai, const float* __restrict__ aj,
                         const unsigned* __restrict__ menc,
                         const float* __restrict__ denom,
                         float* __restrict__ out, int Nn, int H, int C) {
  int HC = H * C;
  int idx = blockIdx.x * blockDim.x + threadIdx.x;
  if (idx >= Nn * HC) return;
  int n = idx / HC;
  int c = idx - n * HC;
  int h = c / C;
  int nh = n * H + h;
  float es = lrelu(ai[nh] + aj[nh]);
  float alpha = __expf(es - fdec(menc[nh])) / (denom[nh] + 1e-16f);
  out[idx] = Hb[idx] * alpha;
}

// ---------------------------------------------------------------------
// Layer-1 scatter: one wave per edge; lane l owns channels 4l..4l+3
// (head l>>2). 512B coalesced b128 read of h[src], 4 fp32 atomics.
// ---------------------------------------------------------------------
__global__ void edge_scatter128(const int* __restrict__ src, const int* __restrict__ dst,
                                const float* __restrict__ Hb,
                                const float* __restrict__ ai, const float* __restrict__ aj,
                                const unsigned* __restrict__ menc,
                                const float* __restrict__ denom,
                                float* __restrict__ out, int E) {
  int wid = (blockIdx.x * blockDim.x + threadIdx.x) >> 5;
  int lane = threadIdx.x & 31;
  if (wid >= E) return;
  int s = src[wid], d = dst[wid];
  int h = lane >> 2;
  int dh = d * 8 + h, sh = s * 8 + h;
  float ev = lrelu(ai[dh] + aj[sh]);
  float alpha = __expf(ev - fdec(menc[dh])) / (denom[dh] + 1e-16f);
  int c0 = lane * 4;
  v4f hv = *(const v4f*)(Hb + (size_t)s * 128 + c0);
  float* op = out + (size_t)d * 128 + c0;
  atomicAdd(op + 0, hv[0] * alpha);
  atomicAdd(op + 1, hv[1] * alpha);
  atomicAdd(op + 2, hv[2] * alpha);
  atomicAdd(op + 3, hv[3] * alpha);
}

// ---------------------------------------------------------------------
// Layer-2 scatter: one wave per edge; lane l owns channels l and l+32.
// ---------------------------------------------------------------------
__global__ void edge_scatter40(const int* __restrict__ src, const int* __restrict__ dst,
                               const float* __restrict__ Hb,
                               const float* __restrict__ ai, const float* __restrict__ aj,
                               const unsigned* __restrict__ menc,
                               const float* __restrict__ denom,
                               float* __restrict__ out, int E) {
  int wid = (blockIdx.x * blockDim.x + threadIdx.x) >> 5;
  int lane = threadIdx.x & 31;
  if (wid >= E) return;
  int s = src[wid], d = dst[wid];
  float ev = lrelu(ai[d] + aj[s]);
  float alpha = __expf(ev - fdec(menc[d])) / (denom[d] + 1e-16f);
  atomicAdd(&out[(size_t)d * 40 + lane], Hb[(size_t)s * 40 + lane] * alpha);
  if (lane < 8) {
    atomicAdd(&out[(size_t)d * 40 + 32 + lane],
              Hb[(size_t)s * 40 + 32 + lane] * alpha);
  }
}

// ---------------------------------------------------------------------
// Layer-1 finalize: + bias, ELU (in place); HC is a power of two (128)
// ---------------------------------------------------------------------
__global__ void bias_elu(float* __restrict__ buf, const float* __restrict__ bias,
                         int total, int HCmask) {
  int idx = blockIdx.x * blockDim.x + threadIdx.x;
  if (idx >= total) return;
  float v = buf[idx] + bias[idx & HCmask];
  buf[idx] = v > 0.f ? v : (__expf(v) - 1.f);
}

// ---------------------------------------------------------------------
// Layer-2 finalize: + bias2, log_softmax over channels (in place)
// ---------------------------------------------------------------------
__global__ void logsoftmax_bias(float* __restrict__ out, const float* __restrict__ bias,
                                int Nn, int Cc) {
  int n = blockIdx.x * blockDim.x + threadIdx.x;
  if (n >= Nn) return;
  float buf[64];
  float mx = -INFINITY;
  for (int c = 0; c < Cc; ++c) {
    float v = out[(size_t)n * Cc + c] + bias[c];
    buf[c] = v;
    mx = fmaxf(mx, v);
  }
  float s = 0.f;
  for (int c = 0; c < Cc; ++c) s += __expf(buf[c] - mx);
  float ls = logf(s) + mx;
  for (int c = 0; c < Cc; ++c) out[(size_t)n * Cc + c] = buf[c] - ls;
}

extern "C" void kernel_launch(void* const* d_in, const int* in_sizes, int n_in,
                              void* d_out, int out_size, void* d_ws, size_t ws_size,
                              hipStream_t stream) {
  const float* x     = (const float*)d_in[0];
  const int*   ei    = (const int*)d_in[1];
  const float* W1    = (const float*)d_in[2];
  const float* b1    = (const float*)d_in[3];
  const float* att1  = (const float*)d_in[4];
  const float* bias1 = (const float*)d_in[5];
  const float* W2    = (const float*)d_in[6];
  const float* b2    = (const float*)d_in[7];
  const float* att2  = (const float*)d_in[8];
  const float* bias2 = (const float*)d_in[9];

  const int IN = 128, H1n = 8, C1n = 16, HC1 = 128, H2n = 1, C2n = 40;
  const int N = in_sizes[0] / IN;
  const int E = in_sizes[1] / 2;
  const int* src = ei;
  const int* dst = ei + E;

  // ---- workspace carve-out ----
  char* ws = (char*)d_ws;
  size_t off = 0;
  auto carve = [&](size_t bytes) -> char* {
    char* p = ws + off;
    off += (bytes + 255) & ~(size_t)255;
    return p;
  };
  float*    h1  = (float*)carve((size_t)N * HC1 * 4);
  float*    o1  = (float*)carve((size_t)N * HC1 * 4);
  float*    ai1 = (float*)carve((size_t)N * H1n * 4);
  float*    aj1 = (float*)carve((size_t)N * H1n * 4);
  unsigned* m1  = (unsigned*)carve((size_t)N * H1n * 4);
  float*    dn1 = (float*)carve((size_t)N * H1n * 4);
  float*    h2  = (float*)carve((size_t)N * C2n * 4);
  float*    ai2 = (float*)carve((size_t)N * 4);
  float*    aj2 = (float*)carve((size_t)N * 4);
  unsigned* m2  = (unsigned*)carve((size_t)N * 4);
  float*    dn2 = (float*)carve((size_t)N * 4);
  float*    o2  = (float*)d_out;  // layer-2 accumulator lives in d_out

  const int B = 256;
  auto nb = [](long long total, int b) { return (int)((total + b - 1) / b); };
  const int mtiles = (N + 15) / 16;

  // ================= Layer 1 =================
  wmma_gemm_lds<<<mtiles, 32 * 8, 0, stream>>>(x, W1, b1, h1, N, IN, HC1);
  attn_scalars<<<nb((long long)N * H1n, B), B, 0, stream>>>(h1, att1, ai1, aj1, m1,
                                                            N, H1n, C1n);
  edge_max8<<<nb(E, B), B, 0, stream>>>(src, dst, ai1, aj1, m1, E);
  denom_init<<<nb((long long)N * H1n, B), B, 0, stream>>>(ai1, aj1, m1, dn1, N * H1n);
  edge_denom8<<<nb(E, B), B, 0, stream>>>(src, dst, ai1, aj1, m1, dn1, E);
  out_init<<<nb((long long)N * HC1, B), B, 0, stream>>>(h1, ai1, aj1, m1, dn1, o1,
                                                        N, H1n, C1n);
  edge_scatter128<<<nb((long long)E * 32, B), B, 0, stream>>>(src, dst, h1, ai1, aj1,
                                                              m1, dn1, o1, E);
  bias_elu<<<nb((long long)N * HC1, B), B, 0, stream>>>(o1, bias1, N * HC1, HC1 - 1);

  // ================= Layer 2 =================
  wmma_gemm_lds<<<mtiles, 32 * 3, 0, stream>>>(o1, W2, b2, h2, N, HC1, C2n);
  attn_scalars<<<nb(N, B), B, 0, stream>>>(h2, att2, ai2, aj2, m2, N, H2n, C2n);
  edge_max<<<nb(E, B), B, 0, stream>>>(src, dst, ai2, aj2, m2, E, H2n);
  denom_init<<<nb(N, B), B, 0, stream>>>(ai2, aj2, m2, dn2, N);
  edge_denom<<<nb(E, B), B, 0, stream>>>(src, dst, ai2, aj2, m2, dn2, E, H2n);
  out_init<<<nb((long long)N * C2n, B), B, 0, stream>>>(h2, ai2, aj2, m2, dn2, o2,
                                                        N, H2n, C2n);
  edge_scatter40<<<nb((long long)E * 32, B), B, 0, stream>>>(src, dst, h2, ai2, aj2,
                                                             m2, dn2, o2, E);
  logsoftmax_bias<<<nb(N, B), B, 0, stream>>>(o2, bias2, N, C2n);
}